// HashEncoderHyFluid_54099408060467
// MI455X (gfx1250) — compile-verified
//
#include <hip/hip_runtime.h>
#include <cmath>
#include <cstdint>
#include <cstddef>

// ---------------------------------------------------------------------------
// HashEncoderHyFluid for MI455X (gfx1250).
//
// Workload shape: 1M points x 16 scales x 16 corners x 2 features gathered
// from a 58MB table -> L2-resident gather kernel (192MB L2). No matmul
// structure (per-point private feature matrices), so WMMA does not apply;
// we use the CDNA5 async global->LDS path for coordinate staging and
// non-temporal b128 stores for the 128MB streaming output so it does not
// evict the table from L2.
// ---------------------------------------------------------------------------

typedef float v2f __attribute__((ext_vector_type(2)));
typedef float v4f __attribute__((ext_vector_type(4)));

struct ScaleParam {
  float    resf[4];   // per-dim resolution (float, for pos scaling)
  uint32_t mult[4];   // {1, st1, st2, st3} dense  |  {1, P1, P2, P3} hashed
  uint32_t mask;      // size-1 (pow2, hashed) | 0xFFFFFFFF (dense, idx<size)
  uint32_t offset;    // float offset of this level's sub-table
  uint32_t hashed;    // 1 = xor combine, 0 = add combine (wave-uniform)
  uint32_t _pad;
};
struct EncCfg { ScaleParam s[16]; };

#define TPB 256
#define NSCALES 16

__global__ __launch_bounds__(TPB)
void hashenc_kernel(const float* __restrict__ xyzt,
                    const float* __restrict__ tbl,
                    float* __restrict__ out,
                    int n, EncCfg cfg)
{
  __shared__ v4f sh[TPB];
  const int tid = threadIdx.x;
  const int p   = blockIdx.x * TPB + tid;
  const int src = (p < n) ? p : (n - 1);

  // Stage this thread's 16B coordinate record through the CDNA5 async
  // global->LDS engine (ASYNCcnt-tracked; no VGPR staging, no barrier
  // needed since each thread reads only its own slot).
  {
    uint32_t lds_off = (uint32_t)(uintptr_t)(&sh[tid]);
    uint64_t gaddr   = (uint64_t)(uintptr_t)(xyzt + (size_t)src * 4);
    asm volatile("global_load_async_to_lds_b128 %0, %1, off"
                 :: "v"(lds_off), "v"(gaddr) : "memory");
    asm volatile("s_wait_asynccnt 0x0" ::: "memory");
  }
  const v4f c4 = sh[tid];
  if (p >= n) return;

  float outv[2 * NSCALES];

#pragma unroll
  for (int s = 0; s < NSCALES; ++s) {
    const ScaleParam sp = cfg.s[s];

    float px = c4.x * sp.resf[0];
    float py = c4.y * sp.resf[1];
    float pz = c4.z * sp.resf[2];
    float pw = c4.w * sp.resf[3];
    float gx = floorf(px), gy = floorf(py), gz = floorf(pz), gw = floorf(pw);
    float fx = px - gx,    fy = py - gy,    fz = pz - gz,    fw = pw - gw;
    uint32_t ux = (uint32_t)(int)gx;
    uint32_t uy = (uint32_t)(int)gy;
    uint32_t uz = (uint32_t)(int)gz;
    uint32_t uw = (uint32_t)(int)gw;

    // Per-dimension lo/hi index terms (mult[0] == 1 for both paths).
    uint32_t tx0 = ux,              tx1 = ux + 1u;
    uint32_t ty0 = uy * sp.mult[1], ty1 = ty0 + sp.mult[1];
    uint32_t tz0 = uz * sp.mult[2], tz1 = tz0 + sp.mult[2];
    uint32_t tw0 = uw * sp.mult[3], tw1 = tw0 + sp.mult[3];

    // Quadrilinear weights via 2x prefix products (8 muls + 16 muls total).
    float wxy[4] = { (1.f-fx)*(1.f-fy), fx*(1.f-fy), (1.f-fx)*fy, fx*fy };
    float wzw[4] = { (1.f-fz)*(1.f-fw), fz*(1.f-fw), (1.f-fz)*fw, fz*fw };

    uint32_t idx[16];
    if (sp.hashed) {            // wave-uniform branch; hashed size is 2^19
#pragma unroll
      for (int c = 0; c < 16; ++c) {
        uint32_t h = ((c & 1) ? tx1 : tx0) ^ ((c & 2) ? ty1 : ty0)
                   ^ ((c & 4) ? tz1 : tz0) ^ ((c & 8) ? tw1 : tw0);
        idx[c] = h & sp.mask;
      }
    } else {                    // dense: idx < raw <= size, no modulo
#pragma unroll
      for (int c = 0; c < 16; ++c) {
        idx[c] = ((c & 1) ? tx1 : tx0) + ((c & 2) ? ty1 : ty0)
               + ((c & 4) ? tz1 : tz0) + ((c & 8) ? tw1 : tw0);
      }
    }

    // Batch of 16 b64 gathers (SGPR base + VGPR offset); keep them
    // temporal so the 58MB table stays resident in the 192MB L2.
    const float* base = tbl + sp.offset;
    v2f fv[16];
#pragma unroll
    for (int c = 0; c < 16; ++c)
      fv[c] = *(const v2f*)(base + 2u * idx[c]);

    float a0 = 0.f, a1 = 0.f;
#pragma unroll
    for (int c = 0; c < 16; ++c) {
      float w = wxy[c & 3] * wzw[c >> 2];
      a0 = fmaf(w, fv[c].x, a0);
      a1 = fmaf(w, fv[c].y, a1);
    }
    outv[2*s]   = a0;
    outv[2*s+1] = a1;
  }

  // Streaming 128B/point output: 8x non-temporal b128 stores; a wave writes
  // 4KB contiguous. NT keeps the hash table from being evicted from L2.
  v4f* ob = (v4f*)(out + (size_t)p * (2 * NSCALES));
#pragma unroll
  for (int k = 0; k < 8; ++k) {
    v4f v = { outv[4*k], outv[4*k+1], outv[4*k+2], outv[4*k+3] };
    __builtin_nontemporal_store(v, ob + k);
  }
}

// ---------------------------------------------------------------------------
// Host: rebuild the reference's static config with the exact numpy double-
// precision formulas (b = exp((ln max - ln min)/15), res = ceil(min*b^s),
// pad-to-8, cap at 2^19). Capped (hashed) levels are provably power-of-two
// sized, dense levels provably need no modulo.
// ---------------------------------------------------------------------------
static void build_cfg(EncCfg& cfg)
{
  const double minr[4] = {16.0, 16.0, 16.0, 16.0};
  const double maxr[4] = {256.0, 256.0, 256.0, 128.0};
  const uint32_t primes[4] = {1u, 2654435761u, 805459861u, 3674653429u};
  double b[4];
  for (int d = 0; d < 4; ++d)
    b[d] = exp((log(maxr[d]) - log(minr[d])) / 15.0);

  long long total = 0;
  for (int s = 0; s < 16; ++s) {
    long long res[4];
    for (int d = 0; d < 4; ++d)
      res[d] = (long long)ceil(minr[d] * pow(b[d], (double)s));
    long long raw = (res[0]+1) * (res[1]+1) * (res[2]+1) * (res[3]+1);
    long long p = (raw % 8 == 0) ? raw : ((raw + 7) / 8) * 8;
    if (p > (1ll << 19)) p = (1ll << 19);
    const bool dense = (raw <= p);

    ScaleParam& sp = cfg.s[s];
    for (int d = 0; d < 4; ++d) sp.resf[d] = (float)res[d];
    if (dense) {
      uint32_t st1 = (uint32_t)(res[0] + 1);
      uint32_t st2 = st1 * (uint32_t)(res[1] + 1);
      uint32_t st3 = st2 * (uint32_t)(res[2] + 1);
      sp.mult[0] = 1u; sp.mult[1] = st1; sp.mult[2] = st2; sp.mult[3] = st3;
      sp.mask   = 0xFFFFFFFFu;
      sp.hashed = 0u;
    } else {
      for (int d = 0; d < 4; ++d) sp.mult[d] = primes[d];
      sp.mask   = (uint32_t)p - 1u;   // capped => p == 2^19 (power of two)
      sp.hashed = 1u;
    }
    sp.offset = (uint32_t)total;
    sp._pad   = 0u;
    total += p * 2;                    // F = 2 features per level
  }
}

extern "C" void kernel_launch(void* const* d_in, const int* in_sizes, int n_in,
                              void* d_out, int out_size, void* d_ws, size_t ws_size,
                              hipStream_t stream)
{
  (void)n_in; (void)out_size; (void)d_ws; (void)ws_size;
  const float* xyzt = (const float*)d_in[0];
  const float* tbl  = (const float*)d_in[1];
  float* out = (float*)d_out;
  const int n = in_sizes[0] / 4;       // (N,4) coordinates

  EncCfg cfg;
  build_cfg(cfg);

  const int blocks = (n + TPB - 1) / TPB;
  hashenc_kernel<<<dim3(blocks), dim3(TPB), 0, stream>>>(xyzt, tbl, out, n, cfg);
}